// NLinear_51539607552763
// MI455X (gfx1250) — compile-verified
//
#include <hip/hip_runtime.h>
#include <hip/hip_bf16.h>

// Problem constants (match reference)
#define Bn 512
#define Wn 512
#define En 512
#define EPSc 1e-5f

typedef __attribute__((ext_vector_type(2))) float v2f;
typedef __attribute__((ext_vector_type(8))) float v8f;

// -----------------------------------------------------------------------------
// Kernel 1: streaming contraction.
//   lin[b,e] = sum_w x[b,w,e]*W1[e,w] + b1[e] - x[b,W-1,e]*sum_w W1[e,w]
//   seq[b,e] = x[b,W-1,e]
// Lanes cover contiguous e -> every w-step the block reads 1KB contiguous of x.
// This kernel moves the 512MB of x exactly once; everything else is noise.
// -----------------------------------------------------------------------------
__global__ __launch_bounds__(256) void k1_lin(const float* __restrict__ x,
                                              const float* __restrict__ W1,
                                              const float* __restrict__ b1,
                                              float* __restrict__ lin,
                                              float* __restrict__ seq) {
  const int e = blockIdx.x * 256 + threadIdx.x;   // 0..511
  const int b = blockIdx.y;                        // 0..511
  const float* xcol  = x  + ((size_t)b * Wn) * En + e;  // stride En over w
  const float* w1row = W1 + (size_t)e * Wn;             // contiguous over w

  float acc = 0.0f;
  float s1  = 0.0f;
  for (int w = 0; w < Wn; w += 4) {
    // prefetch x a few steps ahead (emits global_prefetch_b8 on gfx1250)
    if (w + 16 < Wn) __builtin_prefetch(xcol + (size_t)(w + 16) * En, 0, 0);
    const float4 wv = *(const float4*)(w1row + w);
    const float x0 = xcol[(size_t)(w + 0) * En];
    const float x1 = xcol[(size_t)(w + 1) * En];
    const float x2 = xcol[(size_t)(w + 2) * En];
    const float x3 = xcol[(size_t)(w + 3) * En];
    acc = fmaf(x0, wv.x, acc);
    acc = fmaf(x1, wv.y, acc);
    acc = fmaf(x2, wv.z, acc);
    acc = fmaf(x3, wv.w, acc);
    s1 += (wv.x + wv.y) + (wv.z + wv.w);
  }
  const float sl = xcol[(size_t)(Wn - 1) * En];   // hot in cache (just loaded)
  lin[(size_t)b * En + e] = acc + b1[e] - sl * s1;
  seq[(size_t)b * En + e] = sl;
}

// -----------------------------------------------------------------------------
// Kernel 2: per-channel batch-norm statistics (training mode, biased var),
// folded to an affine:  t = lin*scale + shift  (+ seq_last added later).
// Lanes cover contiguous e, loop over b -> coalesced 1KB reads per step.
// -----------------------------------------------------------------------------
__global__ __launch_bounds__(256) void k2_stats(const float* __restrict__ lin,
                                                const float* __restrict__ gamma,
                                                const float* __restrict__ beta,
                                                float* __restrict__ scale,
                                                float* __restrict__ shift) {
  const int e = blockIdx.x * 256 + threadIdx.x;
  float s = 0.0f, sq = 0.0f;
  for (int b = 0; b < Bn; ++b) {
    const float v = lin[(size_t)b * En + e];
    s += v;
    sq = fmaf(v, v, sq);
  }
  const float mean = s * (1.0f / (float)Bn);
  const float var  = sq * (1.0f / (float)Bn) - mean * mean;  // biased
  const float rstd = rsqrtf(var + EPSc);
  const float sc   = gamma[e] * rstd;
  scale[e] = sc;
  shift[e] = beta[e] - mean * sc;
}

// -----------------------------------------------------------------------------
// Kernel 3: final projection  out[b] = sum_e t[b,e]*W2[e] + b2  via
// V_WMMA_F32_16X16X4_F32. One wave (32 lanes) per 16-row batch tile.
//
// A (16x4, MxK) per ISA layout: lanes 0-15 hold M=lane with {K=0,K=1} in the
// two A VGPRs; lanes 16-31 hold M=lane-16 with {K=2,K=3}.
// B (4x16, KxN): W2[e0+k] broadcast across all N columns (layout mirrors A:
// lanes 0-15 rows K=0/1, lanes 16-31 rows K=2/3).
// D (16x16): every column holds the same dot product; extract column N=0:
// lane 0 VGPRs 0-7 = rows M=0..7, lane 16 VGPRs 0-7 = rows M=8..15.
// -----------------------------------------------------------------------------
__global__ __launch_bounds__(32) void k3_out(const float* __restrict__ lin,
                                             const float* __restrict__ seq,
                                             const float* __restrict__ scale,
                                             const float* __restrict__ shift,
                                             const float* __restrict__ W2,
                                             const float* __restrict__ b2,
                                             float* __restrict__ out) {
  const int lane = threadIdx.x;            // 0..31, full wave32
  const int b0   = blockIdx.x * 16;        // batch tile base
  const int row  = b0 + (lane & 15);       // M for this lane's A elements
  const int koff = (lane >> 4) << 1;       // 0 for lanes 0-15, 2 for 16-31

  const float* lrow = lin + (size_t)row * En;
  const float* srow = seq + (size_t)row * En;

  v8f d = {0.f, 0.f, 0.f, 0.f, 0.f, 0.f, 0.f, 0.f};

  for (int e0 = 0; e0 < En; e0 += 4) {
    const int e = e0 + koff;               // even -> 8B aligned float2 loads
    const v2f lv = *(const v2f*)(lrow + e);
    const v2f sv = *(const v2f*)(srow + e);
    const v2f sc = *(const v2f*)(scale + e);
    const v2f sh = *(const v2f*)(shift + e);

    v2f a;                                 // A-matrix fragment (2 VGPRs)
    a.x = fmaf(lv.x, sc.x, sh.x) + sv.x;
    a.y = fmaf(lv.y, sc.y, sh.y) + sv.y;

    v2f bm;                                // B-matrix fragment: W2 broadcast
    bm.x = W2[e];
    bm.y = W2[e + 1];

    // 8 args: (neg_a, A, neg_b, B, c_mod, C, reuse_a, reuse_b)
    d = __builtin_amdgcn_wmma_f32_16x16x4_f32(false, a, false, bm,
                                              (short)0, d, false, false);
  }

  const float bias = b2[0];
  if (lane == 0) {
#pragma unroll
    for (int i = 0; i < 8; ++i) out[b0 + i] = d[i] + bias;       // M=0..7
  } else if (lane == 16) {
#pragma unroll
    for (int i = 0; i < 8; ++i) out[b0 + 8 + i] = d[i] + bias;   // M=8..15
  }
}

// -----------------------------------------------------------------------------
// Host launcher. Inputs in setup_inputs() order: x, W1, b1, gamma, beta, W2, b2.
// Workspace: lin (1MB) | seq (1MB) | scale (2KB) | shift (2KB).
// -----------------------------------------------------------------------------
extern "C" void kernel_launch(void* const* d_in, const int* in_sizes, int n_in,
                              void* d_out, int out_size, void* d_ws, size_t ws_size,
                              hipStream_t stream) {
  const float* x     = (const float*)d_in[0];
  const float* W1    = (const float*)d_in[1];
  const float* b1    = (const float*)d_in[2];
  const float* gamma = (const float*)d_in[3];
  const float* beta  = (const float*)d_in[4];
  const float* W2    = (const float*)d_in[5];
  const float* b2    = (const float*)d_in[6];
  float* out = (float*)d_out;

  char* ws = (char*)d_ws;
  float* lin   = (float*)(ws);
  float* seq   = (float*)(ws + (size_t)Bn * En * sizeof(float));
  float* scale = (float*)(ws + 2 * (size_t)Bn * En * sizeof(float));
  float* shift = scale + En;

  k1_lin  <<<dim3(En / 256, Bn), 256, 0, stream>>>(x, W1, b1, lin, seq);
  k2_stats<<<dim3(En / 256),     256, 0, stream>>>(lin, gamma, beta, scale, shift);
  k3_out  <<<dim3(Bn / 16),       32, 0, stream>>>(lin, seq, scale, shift, W2, b2, out);
}